// Group_38843684225789
// MI455X (gfx1250) — compile-verified
//
#include <hip/hip_runtime.h>
#include <math.h>

#define B_  16
#define N_  16384
#define G_  512
#define K_  32
#define CH_ 2048
#define D2STRIDE 2049   // 2049 % 64 == 1 -> conflict-free row stride

typedef __attribute__((ext_vector_type(2))) float v2f;
typedef __attribute__((ext_vector_type(8))) float v8f;
typedef __attribute__((ext_vector_type(4))) unsigned int v4u;
typedef __attribute__((ext_vector_type(8))) int v8i;
typedef __attribute__((ext_vector_type(4))) int v4i;

// ---------------------------------------------------------------------------
// Kernel 1: farthest point sampling, one workgroup per batch.
// Entire point cloud (SoA, 192KB) + dist array (64KB) live in CDNA5's 320KB LDS.
// The initial 192KB global->LDS slab copy is issued through the Tensor Data
// Mover (tensor_load_to_lds, TENSORcnt) as a 1D descriptor; a coalesced manual
// copy after s_wait_tensorcnt/barrier is the authoritative writer (TDM field
// semantics are probe-verified for arity only, so it must not be load-bearing).
// 512 iterations; argmax via wave32 shuffle reduce + 1 cross-wave pass.
// Records center coords directly (reference records `far` BEFORE updating dist).
// ---------------------------------------------------------------------------
__global__ void fps_kernel(const float* __restrict__ xyz,
                           float* __restrict__ centerOut) // (B,G,3)
{
    extern __shared__ float smem[];
    float* px   = smem;            // N   (pts slab starts at dynamic-LDS base)
    float* py   = px + N_;         // N
    float* pz   = py + N_;         // N
    float* dist = pz + N_;         // N
    float* rv   = dist + N_;       // 32 (per-wave best value)
    int*   ri   = (int*)(rv + 32); // 32 (per-wave best index)
    int*   sfar = ri + 32;         // 1

    const int tid = threadIdx.x;
    const int b   = blockIdx.x;
    const float* src = xyz + (size_t)b * N_ * 3;

    // ---- TDM: DMA the 49152-float xyz slab into the LDS pts region ----
    if (tid == 0) {
        const unsigned NELT = (unsigned)(N_ * 3);          // 49152 elements
        const unsigned long long ga = (unsigned long long)(uintptr_t)src;
        v4u g0;
        g0.x = 1u;                                         // count=1 (valid user D#)
        g0.y = 0u;                                         // lds_addr = 0 (pts base)
        g0.z = (unsigned)(ga & 0xFFFFFFFFu);               // global_addr[31:0]
        g0.w = (unsigned)((ga >> 32) & 0x01FFFFFFu)        // global_addr[56:32]
             | 0x80000000u;                                // type = 2 ("image")
        v8i g1;
        g1[0] = 0x20000;                                   // data_size=2 (4 bytes)
        g1[1] = (int)((NELT & 0xFFFFu) << 16);             // tensor_dim0[15:0]
        g1[2] = (int)(NELT >> 16);                         // tensor_dim0[31:16]
        g1[3] = (int)((NELT & 0xFFFFu) << 16);             // tile_dim0 (0xC000 fits)
        g1[4] = 0;                                         // tile_dim1/2 unused
        g1[5] = (int)NELT;                                 // tensor_dim0_stride lo
        g1[6] = 0;
        g1[7] = 0;
        v4i gz  = {0, 0, 0, 0};
        v8i gz8 = {0, 0, 0, 0, 0, 0, 0, 0};
        // 6-arg toolchain variant: (g0, g1, g2, g3, g4, cpol)
        __builtin_amdgcn_tensor_load_to_lds(g0, g1, gz, gz, gz8, 0);
        __builtin_amdgcn_s_wait_tensorcnt(0);
    }
    __syncthreads();

    // ---- Authoritative coalesced copy into SoA layout (overwrites slab) ----
    for (int j = tid; j < N_ * 3; j += 1024) {
        float v = src[j];
        int i = j / 3;
        int c = j - 3 * i;
        if (c == 0) px[i] = v; else if (c == 1) py[i] = v; else pz[i] = v;
    }
    for (int i = tid; i < N_; i += 1024) dist[i] = 1e10f;
    if (tid == 0) sfar[0] = 0;     // deterministic_fps: first centroid = point 0
    __syncthreads();

    const int lane = tid & 31;
    const int wave = tid >> 5;

    for (int t = 0; t < G_; ++t) {
        const int far = sfar[0];
        const float cx = px[far], cy = py[far], cz = pz[far];
        if (tid == 0) {
            size_t o = ((size_t)b * G_ + t) * 3;
            centerOut[o + 0] = cx;
            centerOut[o + 1] = cy;
            centerOut[o + 2] = cz;
        }
        // dist = min(dist, ||p - c||^2); track local argmax (first-index ties)
        float bestv = -1.0f; int besti = 0;
        for (int i = tid; i < N_; i += 1024) {
            float dx = px[i] - cx, dy = py[i] - cy, dz = pz[i] - cz;
            float d  = dx * dx + dy * dy + dz * dz;
            float od = dist[i];
            float nd = (d < od) ? d : od;
            dist[i]  = nd;
            if (nd > bestv) { bestv = nd; besti = i; }
        }
        // wave32 shuffle argmax (lowest index on ties)
        for (int off = 16; off > 0; off >>= 1) {
            float ov = __shfl_down(bestv, off, 32);
            int   oi = __shfl_down(besti, off, 32);
            if (ov > bestv || (ov == bestv && oi < besti)) { bestv = ov; besti = oi; }
        }
        if (lane == 0) { rv[wave] = bestv; ri[wave] = besti; }
        __syncthreads();
        if (tid == 0) {
            float bv = rv[0]; int bi = ri[0];
            for (int w = 1; w < 32; ++w) {
                float v = rv[w]; int i2 = ri[w];
                if (v > bv || (v == bv && i2 < bi)) { bv = v; bi = i2; }
            }
            sfar[0] = bi;
        }
        __syncthreads();
    }
}

// ---------------------------------------------------------------------------
// Kernel 2: kNN (top-32 smallest d^2) + gather + recenter.
// One block per (batch, 16-center tile): 512 blocks, 256 threads (8 waves).
// Inner products via v_wmma_f32_16x16x4_f32 with the K=4 slot folding ||p||^2:
//   A row m  = (cx, cy, cz, 1)
//   B col n  = (-2px, -2py, -2pz, ||p||^2)
//   D[m][n]  = d^2(m,n) - ||c_m||^2   (row-constant shift -> same top-k order)
// d^2 chunk staged in LDS (padded stride), stable sorted insertion keeps top-32.
// ---------------------------------------------------------------------------
__global__ void knn_group_kernel(const float* __restrict__ xyz,
                                 const float* __restrict__ centerOut,
                                 float* __restrict__ nbrOut,   // (B,G,K,3)
                                 float* __restrict__ idxOut)   // (B,G,K) as float
{
    extern __shared__ float smem[];
    float* d2buf = smem;                      // 16 * D2STRIDE
    float* tv    = d2buf + 16 * D2STRIDE;     // 16 * K_ (sorted ascending)
    int*   tix   = (int*)(tv + 16 * K_);      // 16 * K_

    const int tid  = threadIdx.x;
    const int lane = tid & 31;
    const int wave = tid >> 5;
    const int half = lane >> 4;   // 0: K=0,1   1: K=2,3
    const int ln   = lane & 15;

    const int b  = blockIdx.x >> 5;   // / (G_/16)
    const int gt = blockIdx.x & 31;
    const int g0 = gt * 16;

    for (int j = tid; j < 16 * K_; j += 256) { tv[j] = 3.4e38f; tix[j] = 0; }

    // A-matrix operand: lane holds center m = ln, K-pair selected by half-wave
    const float* cp = centerOut + ((size_t)b * G_ + g0 + ln) * 3;
    const float ccx = cp[0], ccy = cp[1], ccz = cp[2];
    v2f amat;
    amat.x = half ? ccz  : ccx;
    amat.y = half ? 1.0f : ccy;
    __syncthreads();

    const float* pts = xyz + (size_t)b * N_ * 3;

    for (int c0 = 0; c0 < N_; c0 += CH_) {
        // ---- Phase A: 128 16-point tiles per chunk, 16 per wave ----
        for (int i = 0; i < 16; ++i) {
            const int tile = wave * 16 + i;
            const int n    = c0 + tile * 16 + ln;      // point index (both halves)
            const float* p = pts + (size_t)n * 3;
            const float pxv = p[0], pyv = p[1], pzv = p[2];
            // locality 3 -> WGP-scope prefetch (pulls into all cache levels)
            if (c0 + CH_ < N_) __builtin_prefetch(p + CH_ * 3, 0, 3);
            const float nrm = pxv * pxv + pyv * pyv + pzv * pzv;
            v2f bmat;
            bmat.x = half ? (-2.0f * pzv) : (-2.0f * pxv);
            bmat.y = half ? nrm           : (-2.0f * pyv);
            v8f acc = {0.f, 0.f, 0.f, 0.f, 0.f, 0.f, 0.f, 0.f};
            acc = __builtin_amdgcn_wmma_f32_16x16x4_f32(
                      /*neg_a=*/false, amat, /*neg_b=*/false, bmat,
                      /*c_mod=*/(short)0, acc, /*reuse_a=*/false, /*reuse_b=*/false);
            const int col = tile * 16 + ln;            // D: N = lane%16
            #pragma unroll
            for (int r = 0; r < 8; ++r) {              // D: M = r + 8*(lane/16)
                const int row = r + 8 * half;
                d2buf[row * D2STRIDE + col] = acc[r];
            }
        }
        __syncthreads();
        // ---- Phase B: stable top-32 insertion, one thread per center row ----
        if (tid < 16) {
            const int row = tid;
            float* vrow = tv  + row * K_;
            int*   irow = tix + row * K_;
            float th = vrow[K_ - 1];
            const float* drow = d2buf + row * D2STRIDE;
            for (int j = 0; j < CH_; ++j) {
                const float v = drow[j];
                if (v < th) {                          // strict: ties keep earlier idx
                    int pos = K_ - 1;
                    while (pos > 0 && v < vrow[pos - 1]) {
                        vrow[pos] = vrow[pos - 1];
                        irow[pos] = irow[pos - 1];
                        --pos;
                    }
                    vrow[pos] = v;
                    irow[pos] = c0 + j;
                    th = vrow[K_ - 1];
                }
            }
        }
        __syncthreads();
    }

    // ---- Epilogue: gather neighbors, recenter, emit indices ----
    if (tid < 16) {
        const int row = tid;
        const int g   = g0 + row;
        const float* cc = centerOut + ((size_t)b * G_ + g) * 3;
        const float cx = cc[0], cy = cc[1], cz = cc[2];
        for (int kk = 0; kk < K_; ++kk) {
            const int idx = tix[row * K_ + kk];
            const float* p = pts + (size_t)idx * 3;
            const size_t ob = (((size_t)b * G_ + g) * K_ + kk) * 3;
            nbrOut[ob + 0] = p[0] - cx;
            nbrOut[ob + 1] = p[1] - cy;
            nbrOut[ob + 2] = p[2] - cz;
            idxOut[((size_t)b * G_ + g) * K_ + kk] = (float)idx;
        }
    }
}

// ---------------------------------------------------------------------------
extern "C" void kernel_launch(void* const* d_in, const int* in_sizes, int n_in,
                              void* d_out, int out_size, void* d_ws, size_t ws_size,
                              hipStream_t stream) {
    (void)in_sizes; (void)n_in; (void)out_size; (void)d_ws; (void)ws_size;
    const float* xyz = (const float*)d_in[0];

    float* out    = (float*)d_out;
    float* nbr    = out;                                   // (B,G,K,3)
    float* center = out + (size_t)B_ * G_ * K_ * 3;        // (B,G,3)
    float* gidx   = center + (size_t)B_ * G_ * 3;          // (B,G,K)

    // Kernel 1: 256KB+ dynamic LDS (legal on CDNA5: 320KB/WGP)
    const size_t lds1 = (size_t)(4 * N_ + 32) * sizeof(float) + 33 * sizeof(int);
    (void)hipFuncSetAttribute((const void*)fps_kernel,
                              hipFuncAttributeMaxDynamicSharedMemorySize, (int)lds1);
    fps_kernel<<<dim3(B_), dim3(1024), lds1, stream>>>(xyz, center);

    // Kernel 2
    const size_t lds2 = (size_t)16 * D2STRIDE * sizeof(float)
                      + (size_t)16 * K_ * (sizeof(float) + sizeof(int));
    (void)hipFuncSetAttribute((const void*)knn_group_kernel,
                              hipFuncAttributeMaxDynamicSharedMemorySize, (int)lds2);
    knn_group_kernel<<<dim3(B_ * (G_ / 16)), dim3(256), lds2, stream>>>(
        xyz, center, nbr, gidx);
}